// Assessor_26225070309947
// MI455X (gfx1250) — compile-verified
//
#include <hip/hip_runtime.h>
#include <stdint.h>

#ifndef __has_builtin
#define __has_builtin(x) 0
#endif

#if __has_builtin(__builtin_amdgcn_global_load_async_to_lds_b32)
#define HAVE_ASYNC 1
#else
#define HAVE_ASYNC 0
#endif

#if __has_builtin(__builtin_amdgcn_s_wait_asynccnt)
#define HAVE_WAIT_ASYNC 1
#else
#define HAVE_WAIT_ASYNC 0
#endif

#define BDIM 4
#define DG 192
#define TILE 32
#define PITCH (TILE + 2)                     // 34
#define LAYER_ELEMS (PITCH * PITCH)          // 1156
#define ZCHUNK 24
#define TPB 256
#define LPT ((LAYER_ELEMS + TPB - 1) / TPB)  // 5 async loads per thread per layer
#define NTILES (DG / TILE)                   // 6
#define NZ (DG / ZCHUNK)                     // 8
#define NBLOCKS (BDIM * NTILES * NTILES * NZ)  // 1152

using gas_int = __attribute__((address_space(1))) int;
using las_int = __attribute__((address_space(3))) int;

template <int N>
__device__ __forceinline__ void wait_async() {
#if HAVE_WAIT_ASYNC
  __builtin_amdgcn_s_wait_asynccnt(N);
#elif HAVE_ASYNC
  asm volatile("s_wait_asynccnt %0" ::"i"(N) : "memory");
#endif
}

// Issue exactly LPT async (or fallback sync) copies per thread for one haloed
// 34x34 layer tile. Uniform instruction count per wave so ASYNCcnt waits are
// exact. Out-of-grid halo cells are never written (LDS was pre-zeroed);
// inactive/padding lanes are redirected to a scratch slot with a safe address.
__device__ __forceinline__ void issue_layer(const float* __restrict__ g,
                                            int b, int L, int h0, int w0,
                                            float* __restrict__ lbuf,
                                            float* __restrict__ lscratch,
                                            int tid) {
  const float* layer = g + (size_t)(b * DG + L) * (size_t)(DG * DG);
#pragma unroll
  for (int k = 0; k < LPT; ++k) {
    int e = tid + k * TPB;
    int r = e / PITCH;
    int c = e - r * PITCH;
    int h = h0 - 1 + r;
    int w = w0 - 1 + c;
    bool ok = (e < LAYER_ELEMS) & (h >= 0) & (h < DG) & (w >= 0) & (w < DG);
    const float* gp = ok ? (layer + h * DG + w) : g;             // safe dummy addr
    float* lp = ok ? (lbuf + e) : (lscratch + (tid & 63));       // dummy LDS slot
#if HAVE_ASYNC
    __builtin_amdgcn_global_load_async_to_lds_b32(
        (gas_int*)(uintptr_t)gp,
        (las_int*)(uint32_t)(uintptr_t)lp, 0, 0);
#else
    *lp = ok ? *gp : 0.0f;
#endif
  }
}

__device__ __forceinline__ float block_reduce(float v, float* red, int tid) {
  red[tid] = v;
  __syncthreads();
  for (int s = TPB >> 1; s > 0; s >>= 1) {
    if (tid < s) red[tid] += red[tid + s];
    __syncthreads();
  }
  float r = red[0];
  __syncthreads();
  return r;
}

__global__ void __launch_bounds__(TPB) voxel_stats_kernel(
    const float* __restrict__ vox, float* __restrict__ partial) {
  __shared__ float tile[6][LAYER_ELEMS];  // [0..4] = 5-deep ring, [5] = zeros
  __shared__ float red[TPB];
  __shared__ float scratch[64];

  const int tid = threadIdx.x;
  int t = blockIdx.x;
  const int zi = t % NZ;      t /= NZ;
  const int twi = t % NTILES; t /= NTILES;
  const int thi = t % NTILES; t /= NTILES;
  const int b = t;
  const int z0 = zi * ZCHUNK, z1 = z0 + ZCHUNK;
  const int h0 = thi * TILE, w0 = twi * TILE;

  // Zero all tiles once: zero-layer + out-of-grid halo cells stay 0 forever.
  for (int i = tid; i < 6 * LAYER_ELEMS; i += TPB) (&tile[0][0])[i] = 0.0f;
  __syncthreads();

  float* zb = tile[5];
  if (z0 > 0) issue_layer(vox, b, z0 - 1, h0, w0, tile[(z0 - 1) % 5], scratch, tid);
  issue_layer(vox, b, z0,     h0, w0, tile[z0 % 5],       scratch, tid);
  issue_layer(vox, b, z0 + 1, h0, w0, tile[(z0 + 1) % 5], scratch, tid);

  float pen = 0.0f, exs = 0.0f, tot = 0.0f;
  const int tx = tid & (TILE - 1);
  const int ty = tid >> 5;  // 0..7
  const int cc = 1 + tx;

  for (int d = z0; d < z1; ++d) {
    // Prefetch distance 2: layer d+2 lands while we compute layer d.
    const int L = d + 2;
    if (L <= z1 && L < DG) {
      issue_layer(vox, b, L, h0, w0, tile[L % 5], scratch, tid);
      wait_async<LPT>();   // everything except the batch just issued is done
    } else {
      wait_async<0>();     // drain: guarantees layer d+1 for the tail layers
    }
    __syncthreads();

    const float* prev = (d == 0) ? zb : tile[(d - 1) % 5];
    const float* cur  = tile[d % 5];
    const float* nxt  = (d + 1 < DG) ? tile[(d + 1) % 5] : zb;

#pragma unroll
    for (int j = 0; j < 4; ++j) {
      const int r = 1 + ty + 8 * j;
      const int idx = r * PITCH + cc;
      const float v = cur[idx];
      const float nsum = cur[idx - PITCH] + cur[idx + PITCH] +
                         cur[idx - 1] + cur[idx + 1] +
                         prev[idx] + nxt[idx];
      tot += v;
      exs += fmaxf(6.0f * v - nsum, 0.0f);
      if (d > 0) {
        const float s9 =
            prev[idx - PITCH - 1] + prev[idx - PITCH] + prev[idx - PITCH + 1] +
            prev[idx - 1]         + prev[idx]         + prev[idx + 1] +
            prev[idx + PITCH - 1] + prev[idx + PITCH] + prev[idx + PITCH + 1];
        pen += v * (1.0f - s9 * (1.0f / 9.0f));
      }
    }
    __syncthreads();
  }

  pen = block_reduce(pen, red, tid);
  exs = block_reduce(exs, red, tid);
  tot = block_reduce(tot, red, tid);
  if (tid == 0) {
    partial[3 * blockIdx.x + 0] = pen;
    partial[3 * blockIdx.x + 1] = exs;
    partial[3 * blockIdx.x + 2] = tot;
  }
}

__global__ void __launch_bounds__(TPB) finalize_kernel(
    const float* __restrict__ partial, float* __restrict__ out) {
  __shared__ float red[TPB];
  const int tid = threadIdx.x;
  float pen = 0.0f, exs = 0.0f, tot = 0.0f;
  for (int i = tid; i < NBLOCKS; i += TPB) {
    pen += partial[3 * i + 0];
    exs += partial[3 * i + 1];
    tot += partial[3 * i + 2];
  }
  pen = block_reduce(pen, red, tid);
  exs = block_reduce(exs, red, tid);
  tot = block_reduce(tot, red, tid);
  if (tid == 0) {
    const float inv_size = 1.0f / ((float)BDIM * DG * DG * DG);
    const float overhang = pen * inv_size;
    const float occ = tot * inv_size;
    const float dv = occ - 0.5f;
    out[0] = overhang;
    out[1] = exs / tot - 10.0f * dv * dv;
  }
}

extern "C" void kernel_launch(void* const* d_in, const int* in_sizes, int n_in,
                              void* d_out, int out_size, void* d_ws, size_t ws_size,
                              hipStream_t stream) {
  (void)in_sizes; (void)n_in; (void)out_size; (void)ws_size;
  const float* vox = (const float*)d_in[0];
  float* out = (float*)d_out;
  float* partial = (float*)d_ws;  // NBLOCKS*3 floats = 13.8 KB
  voxel_stats_kernel<<<NBLOCKS, TPB, 0, stream>>>(vox, partial);
  finalize_kernel<<<1, TPB, 0, stream>>>(partial, out);
}